// LigerLMHeadORPO_8701603741902
// MI455X (gfx1250) — compile-verified
//
#include <hip/hip_runtime.h>
#include <hip/hip_bf16.h>

typedef __attribute__((ext_vector_type(2))) float v2f;
typedef __attribute__((ext_vector_type(8))) float v8f;

#define H_DIM   2048
#define V_DIM   32000
#define B2_DIM  8
#define T_DIM   1024
#define NTOK    (B2_DIM * T_DIM)   // 8192
#define IGN_IDX (-100)

#define TOK_TILE 32                 // 32 tokens/WG -> 256KB x tile in LDS
#define NWAVES   8
#define COLS_PER_WAVE 32            // two 16-col subtiles per wave
#define VTILE    (NWAVES * COLS_PER_WAVE)   // 256; V_DIM/VTILE = 125 exact

#define WMMA_F32(A, B, C) \
    __builtin_amdgcn_wmma_f32_16x16x4_f32(false, (A), false, (B), (short)0, (C), false, false)

// ---------------------------------------------------------------------------
// Kernel 1: fused GEMM (f32 WMMA, 2x2 register tiling) + streaming logsumexp.
// One workgroup = 32 tokens; x tile (32x2048 f32 = 256KB) lives in LDS.
// Per-lane online logsumexp state (no cross-lane ops in the V loop);
// one butterfly + LDS merge at the very end.
// ---------------------------------------------------------------------------
__global__ __launch_bounds__(256) void orpo_lse_kernel(
    const float* __restrict__ x, const long long* __restrict__ y,
    const float* __restrict__ W, float* __restrict__ ptok)
{
    __shared__ float xs[TOK_TILE * H_DIM];      // 256 KB
    __shared__ float wmx[NWAVES][TOK_TILE];     // per-wave row max
    __shared__ float wsx[NWAVES][TOK_TILE];     // per-wave row sumexp
    __shared__ float wtg[NWAVES][TOK_TILE];     // per-wave target logit (-inf if absent)
    __shared__ int   ys[TOK_TILE];

    const int tid    = threadIdx.x;
    const int wave   = tid >> 5;
    const int lane   = tid & 31;
    const int laneLo = lane & 15;
    const int hi     = lane >> 4;               // which 16-lane half
    const int tok0   = blockIdx.x * TOK_TILE;
    const float NEG_INF = -__builtin_inff();

    // Stage contiguous 32x2048 f32 x-tile into LDS with b128 traffic.
    {
        const float4* src = (const float4*)(x + (size_t)tok0 * H_DIM);
        float4* dst = (float4*)xs;
        #pragma unroll 4
        for (int i = tid; i < (TOK_TILE * H_DIM) / 4; i += 256) dst[i] = src[i];
    }
    if (tid < TOK_TILE) ys[tid] = (int)y[tok0 + tid];
    __syncthreads();

    // Column assignment within the 256-wide V tile.
    const int col0 = wave * COLS_PER_WAVE + laneLo;
    const int col1 = col0 + 16;

    // Per-lane running state; slot s = rt*8+r covers row m = rt*16 + r + 8*hi.
    // Branchless target capture: compare vbase against (y - col).
    float rm[16], rs[16], tv[16];
    int   yc0[16], yc1[16];
    #pragma unroll
    for (int rt = 0; rt < 2; ++rt)
        #pragma unroll
        for (int r = 0; r < 8; ++r) {
            const int s = rt * 8 + r;
            const int m = rt * 16 + r + 8 * hi;
            rm[s] = NEG_INF; rs[s] = 0.0f; tv[s] = NEG_INF;
            yc0[s] = ys[m] - col0;
            yc1[s] = ys[m] - col1;
        }

    // A fragment bases (ISA 16x4 f32 A layout): row = laneLo, K offset = 2*hi.
    const float* aBase0 = xs + laneLo * H_DIM + 2 * hi;             // rows 0..15
    const float* aBase1 = xs + (16 + laneLo) * H_DIM + 2 * hi;      // rows 16..31

    for (int vbase = 0; vbase < V_DIM; vbase += VTILE) {
        v8f acc00 = {}, acc01 = {}, acc10 = {}, acc11 = {};
        // B fragment bases (B[k][n] = W[n][k]): lane n = laneLo, K offset 2*hi.
        const float* b0p = W + (size_t)(vbase + col0) * H_DIM + 2 * hi;
        const float* b1p = W + (size_t)(vbase + col1) * H_DIM + 2 * hi;

        // Prefetch next V tile's rows toward L2 while we crunch this one.
        if (vbase + VTILE < V_DIM)
            __builtin_prefetch(W + (size_t)(vbase + VTILE + col0) * H_DIM, 0, 1);

        #pragma unroll 4
        for (int k = 0; k < H_DIM; k += 4) {
            v2f a0 = *(const v2f*)(aBase0 + k);
            v2f a1 = *(const v2f*)(aBase1 + k);
            v2f b0 = *(const v2f*)(b0p + k);
            v2f b1 = *(const v2f*)(b1p + k);
            acc00 = WMMA_F32(a0, b0, acc00);
            acc01 = WMMA_F32(a0, b1, acc01);
            acc10 = WMMA_F32(a1, b0, acc10);
            acc11 = WMMA_F32(a1, b1, acc11);
        }

        // Pure per-lane online logsumexp update + branchless target capture.
        #pragma unroll
        for (int rt = 0; rt < 2; ++rt) {
            #pragma unroll
            for (int r = 0; r < 8; ++r) {
                const int s = rt * 8 + r;
                const float l0 = rt ? acc10[r] : acc00[r];
                const float l1 = rt ? acc11[r] : acc01[r];
                tv[s] = (vbase == yc0[s]) ? l0 : tv[s];
                tv[s] = (vbase == yc1[s]) ? l1 : tv[s];
                const float newm = fmaxf(rm[s], fmaxf(l0, l1));
                rs[s] = rs[s] * expf(rm[s] - newm)
                      + expf(l0 - newm) + expf(l1 - newm);
                rm[s] = newm;
            }
        }
    }

    // One-time butterfly merge across the 16 lanes of each half.
    #pragma unroll
    for (int s = 0; s < 16; ++s) {
        float m = rm[s], sm = rs[s], tg = tv[s];
        #pragma unroll
        for (int msk = 1; msk <= 8; msk <<= 1) {
            const float om = __shfl_xor(m,  msk, 16);
            const float os = __shfl_xor(sm, msk, 16);
            const float ot = __shfl_xor(tg, msk, 16);
            const float nm = fmaxf(m, om);
            sm = sm * expf(m - nm) + os * expf(om - nm);
            m  = nm;
            tg = fmaxf(tg, ot);
        }
        rm[s] = m; rs[s] = sm; tv[s] = tg;
    }

    // Publish per-wave row stats.
    if (laneLo == 0) {
        #pragma unroll
        for (int rt = 0; rt < 2; ++rt)
            #pragma unroll
            for (int r = 0; r < 8; ++r) {
                const int m = rt * 16 + r + 8 * hi;
                const int s = rt * 8 + r;
                wmx[wave][m] = rm[s];
                wsx[wave][m] = rs[s];
                wtg[wave][m] = tv[s];
            }
    }
    __syncthreads();

    // Merge across waves; emit per-token logp.
    if (tid < TOK_TILE) {
        const int m = tid;
        float gm = NEG_INF, gt = NEG_INF;
        #pragma unroll
        for (int w = 0; w < NWAVES; ++w) {
            gm = fmaxf(gm, wmx[w][m]);
            gt = fmaxf(gt, wtg[w][m]);
        }
        float gs = 0.0f;
        #pragma unroll
        for (int w = 0; w < NWAVES; ++w) gs += wsx[w][m] * expf(wmx[w][m] - gm);
        const float lse = gm + logf(gs);
        ptok[tok0 + m] = (ys[m] != IGN_IDX) ? (gt - lse) : 0.0f;
    }
}

// ---------------------------------------------------------------------------
// Kernel 2: per-sequence averages + ORPO scalar loss.
// ---------------------------------------------------------------------------
__global__ __launch_bounds__(256) void orpo_final_kernel(
    const float* __restrict__ ptok, const long long* __restrict__ y,
    float* __restrict__ out)
{
    __shared__ float ssum[B2_DIM];
    __shared__ float scnt[B2_DIM];
    const int tid  = threadIdx.x;
    const int seq  = tid >> 5;           // 8 seqs x 32 lanes
    const int lane = tid & 31;

    float s = 0.0f, c = 0.0f;
    for (int t = lane; t < T_DIM; t += 32) {
        const long long yv = y[seq * T_DIM + t];
        if (yv != (long long)IGN_IDX) { s += ptok[seq * T_DIM + t]; c += 1.0f; }
    }
    #pragma unroll
    for (int m = 1; m < 32; m <<= 1) {
        s += __shfl_xor(s, m, 32);
        c += __shfl_xor(c, m, 32);
    }
    if (lane == 0) { ssum[seq] = s; scnt[seq] = c; }
    __syncthreads();

    if (tid == 0) {
        float avg[B2_DIM];
        float chosenSum = 0.0f, chosenCnt = 0.0f;
        #pragma unroll
        for (int b = 0; b < B2_DIM; ++b) {
            avg[b] = ssum[b] / fmaxf(scnt[b], 1.0f);
            if (b < B2_DIM / 2) { chosenSum += ssum[b]; chosenCnt += scnt[b]; }
        }
        const float chosen_nll = -chosenSum / fmaxf(chosenCnt, 1.0f);
        float ratioSum = 0.0f;
        #pragma unroll
        for (int b = 0; b < B2_DIM / 2; ++b) {
            const float ch = avg[b];
            const float rj = avg[b + B2_DIM / 2];
            const float lo = ch - rj - (log1pf(-expf(ch)) - log1pf(-expf(rj)));
            const float ls = (lo >= 0.0f) ? -log1pf(expf(-lo))
                                          : (lo - log1pf(expf(lo)));
            ratioSum += ls;
        }
        out[0] = chosen_nll - 0.1f * ratioSum / (float)(B2_DIM / 2);
    }
}

extern "C" void kernel_launch(void* const* d_in, const int* in_sizes, int n_in,
                              void* d_out, int out_size, void* d_ws, size_t ws_size,
                              hipStream_t stream) {
    (void)in_sizes; (void)n_in; (void)out_size; (void)ws_size;
    const float*     x = (const float*)d_in[0];
    const long long* y = (const long long*)d_in[1];
    const float*     W = (const float*)d_in[2];
    float* ptok = (float*)d_ws;          // 8192 floats of scratch

    orpo_lse_kernel<<<NTOK / TOK_TILE, 256, 0, stream>>>(x, y, W, ptok);
    orpo_final_kernel<<<1, 256, 0, stream>>>(ptok, y, (float*)d_out);
}